// AbsolutePositionExtractor_5978594476425
// MI455X (gfx1250) — compile-verified
//
#include <hip/hip_runtime.h>

typedef __attribute__((ext_vector_type(2))) float v2f;
typedef __attribute__((ext_vector_type(4))) float v4f;

// One row of output = 64 scales * 4 floats = 256 floats = 1024 B.
// Mapping: 256-thread block = 4 rows; tid>>6 = row-in-block, tid&63 = scale s.
// Each wave32 covers scales 0..31 (or 32..63) of one row -> row is
// wave-uniform, and each wave's 32 float4 NT stores are 512 B contiguous.
__global__ __launch_bounds__(256) void geo_feats_kernel(
    const v2f* __restrict__ ll, v4f* __restrict__ out, int B)
{
    const int tid = threadIdx.x;
    const int row = blockIdx.x * 4 + (tid >> 6);
    const int s   = tid & 63;
    if (row >= B) return;

    // gfx1250 global_prefetch_b8, WGP-scope (high locality): the input stream
    // is consumed by this WGP shortly after -> pull into all cache levels.
    if (tid == 0 && (row + 512) < B) {
        __builtin_prefetch((const void*)&ll[row + 512], 0, 3);
    }

    // Row index is uniform across the wave (tid>>6 is constant per wave32).
    // readfirstlane makes that provable -> backend may use s_load_b64
    // (scalar/constant cache), keeping the VMEM pipe free for the stores.
    const int urow = __builtin_amdgcn_readfirstlane(row);
    const v2f p = ll[urow];

    // 2^s via exponent-field construction (s in [0,63] -> biased exp <= 190).
    const float scale   = __int_as_float((s + 127) << 23);
    const float DEG2RAD = 0.017453292519943295f;
    const float c  = DEG2RAD * scale;
    const float la = p.x * c;
    const float lo = p.y * c;

    v4f o;
    o.x = __sinf(la);   // 1/(2pi) pre-scale + v_sin_f32 (trans unit)
    o.y = __cosf(la);
    o.z = __sinf(lo);
    o.w = __cosf(lo);

    // Streaming output, written once, never re-read: non-temporal b128 store.
    __builtin_nontemporal_store(o, &out[(long long)row * 64 + s]);
}

// Zero-fill the mask [B] + positions [B*2] tail with NT 128-bit stores.
__global__ __launch_bounds__(256) void zero_tail_kernel(float* __restrict__ p,
                                                        long long n)
{
    long long i = ((long long)blockIdx.x * blockDim.x + threadIdx.x) * 4;
    if (i + 3 < n) {
        v4f z = {0.f, 0.f, 0.f, 0.f};
        __builtin_nontemporal_store(z, (v4f*)(p + i));
    } else {
        for (; i < n; ++i) p[i] = 0.f;
    }
}

extern "C" void kernel_launch(void* const* d_in, const int* in_sizes, int n_in,
                              void* d_out, int out_size, void* d_ws, size_t ws_size,
                              hipStream_t stream)
{
    const v2f* ll = (const v2f*)d_in[0];
    const int B = in_sizes[0] / 2;          // in_sizes[0] = B*2 flat floats
    float* out = (float*)d_out;

    const int blocks = (B + 3) / 4;         // 4 rows per 256-thread block
    geo_feats_kernel<<<blocks, 256, 0, stream>>>(ll, (v4f*)out, B);

    // Tail region after features [B*256]: mask [B] + positions [B*2], all zero.
    const long long tail = (long long)out_size - (long long)B * 256;
    if (tail > 0) {
        float* tp = out + (long long)B * 256;   // 16B-aligned (B*1024 bytes in)
        const long long groups = (tail + 3) / 4;
        const int zb = (int)((groups + 255) / 256);
        zero_tail_kernel<<<zb, 256, 0, stream>>>(tp, tail);
    }
}